// VectorQuantizer_88476326297838
// MI455X (gfx1250) — compile-verified
//
#include <hip/hip_runtime.h>

typedef float v2f __attribute__((ext_vector_type(2)));
typedef float v8f __attribute__((ext_vector_type(8)));

#define VQ_D        256
#define VQ_NE       1024
#define VQ_NROWS    65536          // 16*4096
#define Z_ELEMS     16777216       // 16*4096*256
#define CHUNK       16             // codebook rows per LDS buffer
#define NCHUNK      (VQ_NE / CHUNK)            // 64
#define CB_STRIDE   260            // padded float stride -> conflict-free ds_load_b64
#define ROWS_PER_BLOCK 128         // 8 waves * 16 rows
#define KSTEPS      64             // 256 / 4 (K per wmma_f32_16x16x4)
#define LOSS_SCALE  (1.25f / 16777216.0f)      // (BETA+1)/numel

// Stage CHUNK codebook rows into an LDS buffer with async global->LDS copies.
// Each of the 256 threads issues 4 global_load_async_to_lds_b128 (ASYNCcnt).
__device__ __forceinline__ void stage_chunk_async(const float* __restrict__ cb,
                                                  float* dstLds, int chunkBase,
                                                  int tid)
{
    const float4* src = (const float4*)(cb + (size_t)chunkBase * VQ_D);
    #pragma unroll
    for (int it = 0; it < (CHUNK * (VQ_D / 4)) / 256; ++it) {   // 4 iters
        int i    = it * 256 + tid;
        int code = i >> 6;                       // 64 float4 per row
        int kq   = i & 63;
        const float4* gaddr = src + code * (VQ_D / 4) + kq;
        // LDS byte address = low 32 bits of the generic pointer (ISA aperture rule)
        unsigned ldsoff = (unsigned)(uintptr_t)(dstLds + code * CB_STRIDE + kq * 4);
        asm volatile("global_load_async_to_lds_b128 %0, %1, off"
                     :: "v"(ldsoff), "v"(gaddr) : "memory");
    }
}

__global__ __launch_bounds__(256) void vq_wmma_f32_kernel(
    const float* __restrict__ z, const float* __restrict__ cb,
    float* __restrict__ zq, float* __restrict__ idxOut,
    float* __restrict__ lossOut)
{
    __shared__ __align__(16) float cb_s[2][CHUNK * CB_STRIDE];  // 33.3 KB
    __shared__ float c2_all[VQ_NE];                             // 4 KB

    const int tid  = threadIdx.x;
    const int wave = tid >> 5;
    const int lane = tid & 31;
    const int half = lane >> 4;      // 0: K pair (0,1); 1: K pair (2,3)
    const int mrow = lane & 15;      // row (A) / col (B) within 16-tile
    const int rowBase = blockIdx.x * ROWS_PER_BLOCK + wave * 16;

    // ---- kick off async staging of chunk 0 (overlaps the c2 prologue)
    stage_chunk_async(cb, &cb_s[0][0], 0, tid);

    // ---- all 1024 code squared-norms, once per block (4 codes per thread)
    #pragma unroll
    for (int r = 0; r < 4; ++r) {
        int code = r * 256 + tid;
        const float4* p = (const float4*)(cb + (size_t)code * VQ_D);
        float s = 0.0f;
        #pragma unroll 8
        for (int k = 0; k < VQ_D / 4; ++k) {
            float4 v = p[k];
            s += v.x * v.x + v.y * v.y + v.z * v.z + v.w * v.w;
        }
        c2_all[code] = s;
    }

    // ---- preload A fragments: A = -2 * z_tile (16x256), fp32 WMMA layout
    const float* zr = z + (size_t)(rowBase + mrow) * VQ_D + 2 * half;
    v2f aReg[KSTEPS];
    #pragma unroll
    for (int kk = 0; kk < KSTEPS; ++kk) {
        v2f t = *(const v2f*)(zr + 4 * kk);
        aReg[kk] = t * (-2.0f);
    }

    float minval[8];
    int   minidx[8];
    #pragma unroll
    for (int j = 0; j < 8; ++j) { minval[j] = 3.4e38f; minidx[j] = 0; }

    asm volatile("s_wait_asynccnt 0" ::: "memory");
    __syncthreads();

    // ---- double-buffered pipeline: stage chunk n+1 async while WMMA'ing chunk n
    #pragma unroll 1
    for (int chunk = 0; chunk < NCHUNK; ++chunk) {
        const int cur = chunk & 1;
        if (chunk + 1 < NCHUNK)
            stage_chunk_async(cb, &cb_s[cur ^ 1][0], (chunk + 1) * CHUNK, tid);

        // one 16-code tile: fp32 WMMA over K=256
        const float* bp = &cb_s[cur][mrow * CB_STRIDE + 2 * half];
        v8f acc = {};
        #pragma unroll
        for (int kk = 0; kk < KSTEPS; ++kk) {
            v2f b = *(const v2f*)(bp + 4 * kk);
            acc = __builtin_amdgcn_wmma_f32_16x16x4_f32(
                /*neg_a=*/false, aReg[kk], /*neg_b=*/false, b,
                /*c_mod=*/(short)0, acc, /*reuse_a=*/false, /*reuse_b=*/false);
        }
        // acc[j] = -2*dot(z[row], cb[code]); row = j + 8*half, code col = mrow
        const int   code = chunk * CHUNK + mrow;
        const float c2v  = c2_all[code];
        #pragma unroll
        for (int j = 0; j < 8; ++j) {
            float d = c2v + acc[j];
            bool better = d < minval[j];
            minval[j] = better ? d : minval[j];
            minidx[j] = better ? code : minidx[j];
        }

        asm volatile("s_wait_asynccnt 0" ::: "memory");
        __syncthreads();
    }

    // ---- argmin across the 16 lanes of each half (tie -> lowest index)
    #pragma unroll
    for (int j = 0; j < 8; ++j) {
        float mv = minval[j];
        int   mi = minidx[j];
        #pragma unroll
        for (int s = 1; s < 16; s <<= 1) {
            float ov = __shfl_xor(mv, s, 32);
            int   oi = __shfl_xor(mi, s, 32);
            if (ov < mv || (ov == mv && oi < mi)) { mv = ov; mi = oi; }
        }
        minval[j] = mv;
        minidx[j] = mi;
    }

    // ---- gather z_q = cb[idx], write indices, accumulate (z_q - z)^2
    float lsum = 0.0f;
    #pragma unroll
    for (int m = 0; m < 16; ++m) {
        const int srcLane = (m < 8) ? 0 : 16;     // half holding row m's result
        const int c   = __shfl(minidx[m & 7], srcLane, 32);
        const int row = rowBase + m;
        const float4* crow = (const float4*)(cb + (size_t)c * VQ_D);
        const float4* zrow = (const float4*)(z + (size_t)row * VQ_D);
        float4*       qrow = (float4*)(zq + (size_t)row * VQ_D);
        #pragma unroll
        for (int t = 0; t < 2; ++t) {
            int e = lane + 32 * t;
            float4 cv = crow[e];
            float4 zv = zrow[e];
            qrow[e] = cv;
            float dx = cv.x - zv.x, dy = cv.y - zv.y;
            float dz2 = cv.z - zv.z, dw = cv.w - zv.w;
            lsum += dx * dx + dy * dy + dz2 * dz2 + dw * dw;
        }
        if (lane == 0) idxOut[row] = (float)c;
    }

    // ---- loss: wave reduce, one atomic per wave into pre-zeroed slot
    #pragma unroll
    for (int s = 1; s < 32; s <<= 1) lsum += __shfl_xor(lsum, s, 32);
    if (lane == 0) atomicAdd(lossOut, lsum * LOSS_SCALE);
}

extern "C" void kernel_launch(void* const* d_in, const int* in_sizes, int n_in,
                              void* d_out, int out_size, void* d_ws, size_t ws_size,
                              hipStream_t stream) {
    const float* z  = (const float*)d_in[0];   // [16,4096,256] f32
    const float* cb = (const float*)d_in[1];   // [1024,256]  f32
    float* out  = (float*)d_out;
    float* zq   = out;                // 16,777,216 floats
    float* loss = out + Z_ELEMS;      // 1 float
    float* idx  = out + Z_ELEMS + 1;  // 65,536 floats (indices as float)

    hipMemsetAsync(loss, 0, sizeof(float), stream);   // capture-safe memset node

    dim3 grid(VQ_NROWS / ROWS_PER_BLOCK);             // 512 blocks
    dim3 block(256);                                  // 8 wave32
    vq_wmma_f32_kernel<<<grid, block, 0, stream>>>(z, cb, zq, idx, loss);
}